// CrossEntropyLossByFrequencyTier_39977555591279
// MI455X (gfx1250) — compile-verified
//
#include <hip/hip_runtime.h>
#include <hip/hip_bf16.h>
#include <math.h>

// ---------------- problem constants (from reference) ----------------
#define N_ROWS   4096
#define VOCAB_SZ 50257
#define NUM_TIERS 4

constexpr int   NV4      = VOCAB_SZ / 4;             // 12564 full float4s
constexpr int   WAVES    = 8;                        // 256 threads / wave32
constexpr int   PER_WAVE = (NV4 + WAVES - 1) / WAVES; // 1571 vec4 per wave
constexpr float L2E      = 1.4426950408889634f;      // log2(e)
constexpr float LN2      = 0.6931471805599453f;      // ln(2)

// ---------------- CDNA5 async global->LDS path ----------------
#if defined(__has_builtin)
# if __has_builtin(__builtin_amdgcn_global_load_async_to_lds_b128) && \
     __has_builtin(__builtin_amdgcn_s_wait_asynccnt)
#  define HAVE_ASYNC 1
# endif
#endif
#ifndef HAVE_ASYNC
# define HAVE_ASYNC 0
#endif

#if defined(__HIP_DEVICE_COMPILE__)
# if HAVE_ASYNC
#  warning "CDNA5: async global_load_async_to_lds_b128 path ENABLED"
# else
#  warning "CDNA5: async builtins NOT found -> direct-load fallback"
# endif
#endif

// exact pointee type per hipcc diagnostic: 'int __attribute__((vector_size(16)))'
typedef int v4i_t __attribute__((vector_size(4 * sizeof(int))));
typedef __attribute__((address_space(1))) v4i_t* gptr_b128;
typedef __attribute__((address_space(3))) v4i_t* lptr_b128;

__device__ __forceinline__ void async_cp_b128(const float4* g, float4* l) {
#if HAVE_ASYNC
  __builtin_amdgcn_global_load_async_to_lds_b128(
      (gptr_b128)(v4i_t*)g,
      (lptr_b128)(v4i_t*)l,
      /*offset=*/0, /*cpol=*/0);
#else
  *l = *g;   // synchronous staging fallback (still correct)
#endif
}

#if HAVE_ASYNC
# define WAIT_ASYNC(n) __builtin_amdgcn_s_wait_asynccnt(n)
#else
# define WAIT_ASYNC(n) ((void)0)
#endif

__device__ __forceinline__ float fexp2(float x) {
#if defined(__has_builtin) && __has_builtin(__builtin_amdgcn_exp2f)
  return __builtin_amdgcn_exp2f(x);   // v_exp_f32
#else
  return exp2f(x);
#endif
}
__device__ __forceinline__ float flog2(float x) {
#if defined(__has_builtin) && __has_builtin(__builtin_amdgcn_logf)
  return __builtin_amdgcn_logf(x);    // v_log_f32 (log2)
#else
  return log2f(x);
#endif
}

// combine two (max, sum) online-logsumexp states (exp2 domain)
__device__ __forceinline__ void lse_combine(float& m, float& s, float m2, float s2) {
  float nm = fmaxf(m, m2);
  s = s * fexp2(m - nm) + s2 * fexp2(m2 - nm);
  m = nm;
}

// ---------------- kernel 1: per-row logsumexp - picked logit ----------------
__global__ __launch_bounds__(256)
void row_lse_kernel(const float* __restrict__ inputs,
                    const int*   __restrict__ labels,
                    float*       __restrict__ losses) {
  const int row  = blockIdx.x;
  const int tid  = threadIdx.x;
  const int wid  = tid >> 5;
  const int lane = tid & 31;

  const float*  rowp = inputs + (size_t)row * VOCAB_SZ;
  const float4* gsrc = (const float4*)rowp;

  // 4-deep async ring buffer per wave: 8 waves * 4 slots * 32 lanes * 16B = 16 KB
  __shared__ float4 buf[WAVES][4][32];
  __shared__ float  red_m[WAVES];
  __shared__ float  red_s[WAVES];

  const int base    = wid * PER_WAVE;
  const int end     = min(NV4, base + PER_WAVE);
  const int nchunks = (end - base + 31) >> 5;     // 49 or 50

  // --- prologue: issue 3 chunks ahead ---
  {
    int i0 = base + 0 * 32 + lane; if (i0 < end) async_cp_b128(gsrc + i0, &buf[wid][0][lane]);
    int i1 = base + 1 * 32 + lane; if (i1 < end) async_cp_b128(gsrc + i1, &buf[wid][1][lane]);
    int i2 = base + 2 * 32 + lane; if (i2 < end) async_cp_b128(gsrc + i2, &buf[wid][2][lane]);
  }

  float m = -3.0e38f;   // running max (exp2 domain); avoids -inf NaN traps
  float s = 0.0f;       // running sum of exp2(x*log2e - m)

  for (int i = 0; i < nchunks; ++i) {
    if (i + 3 < nchunks) {                       // wave-uniform branch
      int ii = base + (i + 3) * 32 + lane;
      if (ii < end) async_cp_b128(gsrc + ii, &buf[wid][(i + 3) & 3][lane]);
      WAIT_ASYNC(3);                             // chunk i is now resident
    } else {
      WAIT_ASYNC(0);                             // drain tail
    }

    int idx = base + i * 32 + lane;
    if (idx < end) {
      float4 v  = buf[wid][i & 3][lane];
      float a0 = v.x * L2E, a1 = v.y * L2E, a2 = v.z * L2E, a3 = v.w * L2E;
      float cm = fmaxf(fmaxf(a0, a1), fmaxf(a2, a3));
      float nm = fmaxf(m, cm);
      s = s * fexp2(m - nm)
        + ((fexp2(a0 - nm) + fexp2(a1 - nm)) + (fexp2(a2 - nm) + fexp2(a3 - nm)));
      m = nm;
    }
  }

  // scalar tail elements (VOCAB_SZ % 4 == 1 -> one element, handled by thread 0)
  for (int j = NV4 * 4 + tid; j < VOCAB_SZ; j += 256) {
    float a  = rowp[j] * L2E;
    float nm = fmaxf(m, a);
    s = s * fexp2(m - nm) + fexp2(a - nm);
    m = nm;
  }

  // --- wave32 shuffle reduction of (m, s) ---
  #pragma unroll
  for (int off = 16; off > 0; off >>= 1) {
    float m2 = __shfl_xor(m, off, 32);
    float s2 = __shfl_xor(s, off, 32);
    lse_combine(m, s, m2, s2);
  }
  if (lane == 0) { red_m[wid] = m; red_s[wid] = s; }
  __syncthreads();

  if (tid == 0) {
    float M = red_m[0], S = red_s[0];
    #pragma unroll
    for (int w = 1; w < WAVES; ++w) lse_combine(M, S, red_m[w], red_s[w]);
    float logz   = (M + flog2(S)) * LN2;         // back to natural log
    int   lab    = labels[row];
    float picked = rowp[lab];
    losses[row]  = logz - picked;
  }
}

// ---------------- kernel 2: deterministic tier segment-sum ----------------
__global__ __launch_bounds__(256)
void finalize_kernel(const float* __restrict__ losses,
                     const int*   __restrict__ labels,
                     float*       __restrict__ out) {
  __shared__ float acc[256][8];
  float v[8] = {0, 0, 0, 0, 0, 0, 0, 0};

  for (int r = threadIdx.x; r < N_ROWS; r += 256) {
    int lab = labels[r];
    int t   = (lab >= 100) + (lab >= 1000) + (lab >= 10000);
    v[t]     += losses[r];
    v[4 + t] += 1.0f;
  }
  #pragma unroll
  for (int j = 0; j < 8; ++j) acc[threadIdx.x][j] = v[j];
  __syncthreads();

  for (int off = 128; off > 0; off >>= 1) {
    if (threadIdx.x < off) {
      #pragma unroll
      for (int j = 0; j < 8; ++j) acc[threadIdx.x][j] += acc[threadIdx.x + off][j];
    }
    __syncthreads();
  }

  if (threadIdx.x < NUM_TIERS) {
    out[threadIdx.x] = acc[0][threadIdx.x];                 // values (empty -> 0)
    float c = acc[0][4 + threadIdx.x];
    out[4 + threadIdx.x] = (c == 0.0f) ? 1.0f : c;          // counts (empty -> 1)
  }
}

// ---------------- launcher ----------------
extern "C" void kernel_launch(void* const* d_in, const int* in_sizes, int n_in,
                              void* d_out, int out_size, void* d_ws, size_t ws_size,
                              hipStream_t stream) {
  const float* inputs = (const float*)d_in[0];   // [N_ROWS, VOCAB_SZ] fp32
  const int*   labels = (const int*)d_in[1];     // [N_ROWS] int32
  float*       out    = (float*)d_out;           // 8 floats: values[4], counts[4]
  float*       losses = (float*)d_ws;            // N_ROWS floats scratch (16 KB)

  row_lse_kernel<<<N_ROWS, 256, 0, stream>>>(inputs, labels, losses);
  finalize_kernel<<<1, 256, 0, stream>>>(losses, labels, out);
}